// SparseMultiheadAttention_70944269795742
// MI455X (gfx1250) — compile-verified
//
#include <hip/hip_runtime.h>
#include <cfloat>
#include <cmath>

// CDNA5 (gfx1250) wave32 WMMA types
typedef __attribute__((ext_vector_type(2))) float v2f;
typedef __attribute__((ext_vector_type(8))) float v8f;

// GCC-style vector to match the async-to-LDS builtin's parameter type
// (diagnostic: '__attribute__((__vector_size__(4 * sizeof(int)))) int *')
typedef int v4i_gcc __attribute__((vector_size(16)));
typedef __attribute__((address_space(3))) v4i_gcc lds_v4i;

#define LDS_STRIDE 260  // 256 floats + 4-float pad -> conflict-free ds_load_b64 across 16 rows

__device__ __forceinline__ void async_copy16(const void* gsrc, void* ldst) {
#if __has_builtin(__builtin_amdgcn_global_load_async_to_lds_b128)
  // (src, dst, imm offset, imm cpol); dst is LDS: AS(3) pointer via the integer
  // route (low 32 bits of a generic LDS pointer are the LDS byte offset).
  __builtin_amdgcn_global_load_async_to_lds_b128(
      (v4i_gcc*)gsrc, (lds_v4i*)(unsigned)(uintptr_t)ldst, 0, 0);
#else
  const unsigned l = (unsigned)(uintptr_t)ldst;
  asm volatile("global_load_async_to_lds_b128 %0, %1, off"
               :: "v"(l), "v"(gsrc) : "memory");
#endif
}
__device__ __forceinline__ void wait_async0() {
#if __has_builtin(__builtin_amdgcn_s_wait_asynccnt)
  __builtin_amdgcn_s_wait_asynccnt(0);
#else
  asm volatile("s_wait_asynccnt 0x0" ::: "memory");
#endif
}

__device__ __forceinline__ unsigned flip_f32(float f) {
  unsigned u = __float_as_uint(f);
  return (u & 0x80000000u) ? ~u : (u | 0x80000000u);  // order-preserving float -> uint
}
__device__ __forceinline__ float unflip_f32(unsigned v) {
  return (v & 0x80000000u) ? __uint_as_float(v ^ 0x80000000u) : __uint_as_float(~v);
}

// K1: per block = 16 edges x 1 batch. Async-gather q/k rows to LDS (ASYNCcnt),
// per head diag(Q16x32 * K32x16^T) via two interleaved chains of
// V_WMMA_F32_16X16X4_F32, scale, store logits, global max via flipped-uint atomics.
__global__ __launch_bounds__(128) void sma_logits_wmma(
    const float* __restrict__ q, const float* __restrict__ kmat,
    const long long* __restrict__ e, float* __restrict__ logits,
    unsigned* __restrict__ gmax, int n, int m) {
  __shared__ float qs[16][LDS_STRIDE];
  __shared__ float ks[16][LDS_STRIDE];
  __shared__ int se0[16], se1[16];
  __shared__ unsigned bmax;

  const int tid = threadIdx.x;
  const int bi = blockIdx.y;
  const long long base = (long long)blockIdx.x * 16;

  if (tid < 16) {
    long long ei = base + tid;
    int v0 = 0, v1 = 0;
    if (ei < m) { v0 = (int)e[ei]; v1 = (int)e[(long long)m + ei]; }
    se0[tid] = v0; se1[tid] = v1;
  }
  if (tid == 0) bmax = 0u;
  __syncthreads();

  // Async gather memory->LDS (no VGPR staging): 16 rows x 64 float4 per matrix,
  // consecutive threads -> consecutive float4 within a row (coalesced 1KB rows,
  // L2-resident).  16 async b128 ops in flight per wave, then one wait+barrier.
  const size_t bOff = (size_t)bi * (size_t)n;
  #pragma unroll
  for (int i = tid; i < 16 * 64; i += 128) {
    const int row = i >> 6, c4 = i & 63;
    const float* qsrc = q    + (bOff + (size_t)se0[row]) * 256 + c4 * 4;
    const float* ksrc = kmat + (bOff + (size_t)se1[row]) * 256 + c4 * 4;
    async_copy16(qsrc, &qs[row][c4 * 4]);
    async_copy16(ksrc, &ks[row][c4 * 4]);
  }
  wait_async0();
  __syncthreads();

  const int wave = tid >> 5;
  const int lane = tid & 31;
  const int row  = lane & 15;            // M for A, N for B (edge-in-tile)
  const int koff = (lane >> 4) << 1;     // lanes 16..31 supply K+2, K+3

  float lmax = -FLT_MAX;

  #pragma unroll
  for (int hh = 0; hh < 2; ++hh) {
    const int h = wave * 2 + hh;         // 4 waves x 2 = 8 heads
    const float* qrow = &qs[row][h * 32 + koff];
    const float* krow = &ks[row][h * 32 + koff];
    v8f c0 = {}, c1 = {};                // two chains -> 2x ILP in the XDL pipe
    #pragma unroll
    for (int kc = 0; kc < 8; kc += 2) {  // K = 32 per head, 4 per WMMA
      v2f a0, b0, a1, b1;
      a0.x = qrow[kc * 4];     a0.y = qrow[kc * 4 + 1];   // 8B-aligned ds_load_b64
      b0.x = krow[kc * 4];     b0.y = krow[kc * 4 + 1];
      a1.x = qrow[kc * 4 + 4]; a1.y = qrow[kc * 4 + 5];
      b1.x = krow[kc * 4 + 4]; b1.y = krow[kc * 4 + 5];
      // identical lane->K-slot mapping for A and B keeps the diagonal exact
      // under any common K permutation.
      c0 = __builtin_amdgcn_wmma_f32_16x16x4_f32(false, a0, false, b0, (short)0, c0, false, false);
      c1 = __builtin_amdgcn_wmma_f32_16x16x4_f32(false, a1, false, b1, (short)0, c1, false, false);
    }
    // diag(j): j<8 -> VGPR j @ lane j ; j>=8 -> VGPR j-8 @ lane j+16
    const bool lo = lane < 8, hi = lane >= 24;
    if (lo || hi) {
      const int idx  = lo ? lane : (lane - 24);
      const int edge = lo ? lane : (lane - 16);
      float dv = c0[0] + c1[0];
      #pragma unroll
      for (int t = 1; t < 8; ++t) if (idx == t) dv = c0[t] + c1[t];
      const long long ei = base + edge;
      if (ei < m) {
        const float val = dv * 0.0625f;  // 1/sqrt(KEY_DIM=256)
        logits[((size_t)bi * (size_t)m + (size_t)ei) * 8 + h] = val;
        lmax = fmaxf(lmax, val);
      }
    }
  }

  // wave -> block -> global max (monotone flipped-uint atomics)
  #pragma unroll
  for (int o = 16; o > 0; o >>= 1) lmax = fmaxf(lmax, __shfl_xor(lmax, o, 32));
  if (lane == 0) atomicMax(&bmax, flip_f32(lmax));
  __syncthreads();
  if (tid == 0) atomicMax(gmax, bmax);
}

// K2: ex = exp(logit - gmax) in-place on d_out; atomicAdd into per-(segment,b,head) sums.
__global__ __launch_bounds__(256) void sma_exp_seg(
    float* __restrict__ io, const long long* __restrict__ r,
    float* __restrict__ seg, const unsigned* __restrict__ gmax, int m) {
  const int edge = blockIdx.x * 256 + threadIdx.x;
  const int bi = blockIdx.y;
  if (edge >= m) return;
  const float amax = unflip_f32(*gmax);
  const size_t off = ((size_t)bi * (size_t)m + (size_t)edge) * 8;
  float4 v0 = *(const float4*)(io + off);
  float4 v1 = *(const float4*)(io + off + 4);
  float ex[8] = {expf(v0.x - amax), expf(v0.y - amax), expf(v0.z - amax), expf(v0.w - amax),
                 expf(v1.x - amax), expf(v1.y - amax), expf(v1.z - amax), expf(v1.w - amax)};
  *(float4*)(io + off)     = make_float4(ex[0], ex[1], ex[2], ex[3]);
  *(float4*)(io + off + 4) = make_float4(ex[4], ex[5], ex[6], ex[7]);
  const size_t sb = ((size_t)r[edge] * (size_t)gridDim.y + (size_t)bi) * 8;
  #pragma unroll
  for (int h = 0; h < 8; ++h) atomicAdd(&seg[sb + h], ex[h]);
}

// K3: out = ex / (seg[r] + 1e-16)
__global__ __launch_bounds__(256) void sma_div(
    float* __restrict__ io, const long long* __restrict__ r,
    const float* __restrict__ seg, int m) {
  const int edge = blockIdx.x * 256 + threadIdx.x;
  const int bi = blockIdx.y;
  if (edge >= m) return;
  const size_t off = ((size_t)bi * (size_t)m + (size_t)edge) * 8;
  const size_t sb  = ((size_t)r[edge] * (size_t)gridDim.y + (size_t)bi) * 8;
  float4 v0 = *(const float4*)(io + off);
  float4 v1 = *(const float4*)(io + off + 4);
  const float4 d0 = *(const float4*)(seg + sb);
  const float4 d1 = *(const float4*)(seg + sb + 4);
  v0.x /= (d0.x + 1e-16f); v0.y /= (d0.y + 1e-16f);
  v0.z /= (d0.z + 1e-16f); v0.w /= (d0.w + 1e-16f);
  v1.x /= (d1.x + 1e-16f); v1.y /= (d1.y + 1e-16f);
  v1.z /= (d1.z + 1e-16f); v1.w /= (d1.w + 1e-16f);
  *(float4*)(io + off)     = v0;
  *(float4*)(io + off + 4) = v1;
}

extern "C" void kernel_launch(void* const* d_in, const int* in_sizes, int n_in,
                              void* d_out, int out_size, void* d_ws, size_t ws_size,
                              hipStream_t stream) {
  (void)n_in; (void)ws_size;
  const float*     q = (const float*)d_in[0];
  const float*     k = (const float*)d_in[1];
  const long long* e = (const long long*)d_in[2];  // int64 (2, m)
  const long long* r = (const long long*)d_in[3];  // int64 (m,)
  float* out = (float*)d_out;

  const int m = in_sizes[3];
  const int b = out_size / (m * 8);          // out = (b, m, 8)
  const int n = in_sizes[0] / (b * 256);     // q = (b, n, 256)

  // workspace: [0..63] flipped-uint global max slot; [64..] segment sums (n*b*8 f32)
  unsigned* gmax = (unsigned*)d_ws;
  float* seg = (float*)((char*)d_ws + 64);
  const size_t initBytes = 64 + (size_t)n * (size_t)b * 8 * sizeof(float);
  (void)hipMemsetAsync(d_ws, 0, initBytes, stream);

  dim3 g1((m + 15) / 16, b), b1(128);
  sma_logits_wmma<<<g1, b1, 0, stream>>>(q, k, e, out, gmax, n, m);

  dim3 g2((m + 255) / 256, b), b2(256);
  sma_exp_seg<<<g2, b2, 0, stream>>>(out, r, seg, gmax, m);
  sma_div<<<g2, b2, 0, stream>>>(out, r, seg, m);
}